// Conv2dCQ_39230231281863
// MI455X (gfx1250) — compile-verified
//
#include <hip/hip_runtime.h>
#include <hip/hip_bf16.h>

typedef float v2f __attribute__((ext_vector_type(2)));
typedef float v4f __attribute__((ext_vector_type(4)));
typedef float v8f __attribute__((ext_vector_type(8)));

#define N_IMG 16
#define F_OUT 64
#define H_IN 256
#define W_IN 256
#define H_OUT 254
#define W_OUT 254
#define WTILES 16               // ceil(254/16)
#define WAVES_PER_BLOCK 8

// One wave computes a 64(filter) x 16(w) output tile at a fixed (n, h) using
// v_wmma_f32_16x16x4_f32:  A = weights (M=16 filters x K=4), B = patches
// (K=4 x N=16 w-positions), 3 chained K-chunks (K=9 padded to 12), with the
// bias folded into the first WMMA's C operand.
__global__ __launch_bounds__(256) void conv2dcq_wmma_kernel(
    const float* __restrict__ x,      // (16,3,256,256), only channel 0 used
    const float* __restrict__ wgt,    // (64,3,3,3), only channel 2 used
    const float* __restrict__ bias,   // (64,)
    float* __restrict__ out)          // (16,64,254,254)
{
    const int lane = threadIdx.x & 31;
    const int waveInBlk = threadIdx.x >> 5;
    const int gw = blockIdx.x * WAVES_PER_BLOCK + waveInBlk;

    const int wtile = gw & (WTILES - 1);
    const int h = (gw >> 4) % H_OUT;
    const int n = gw / (WTILES * H_OUT);

    const int w0 = wtile * 16;
    const int half = lane >> 4;     // which K-pair of the 4-wide chunk we hold
    const int lid = lane & 15;

    // ---- B matrix: image patches for 16 consecutive w positions -----------
    // patch[m][k] = x[n, 0, h + k/3, w0 + m + k%3]; lane lid owns column m=lid.
    // Flat offsets for k=0..8: {0,1,2, 256,257,258, 512,513,514}.
    // Per-half offset selection is a v_cndmask, not a branch; the k=8 element
    // is loaded by all lanes (always in-bounds) and zeroed in the upper half.
    const float* xb = x + (size_t)n * (3 * H_IN * W_IN) + (size_t)h * W_IN + w0 + lid;
    const int o00 = half ? 2   : 0;     // chunk0: k = {0,1} | {2,3}
    const int o01 = half ? 256 : 1;
    const int o10 = half ? 512 : 257;   // chunk1: k = {4,5} | {6,7}
    const int o11 = half ? 513 : 258;
    v2f B0, B1, B2;
    B0.x = xb[o00];  B0.y = xb[o01];
    B1.x = xb[o10];  B1.y = xb[o11];
    const float x8 = xb[514];           // k=8 element
    B2.x = half ? 0.0f : x8;
    B2.y = 0.0f;

    // ---- 4 filter tiles of 16: A = w3[f][k] = wgt[f*27 + 18 + k] ----------
    const int ka = half ? 2 : 0;        // chunk0 K base (adjacent pair)
    const int kb2 = half ? 6 : 4;       // chunk1 K base (adjacent pair)
    const int wcol = w0 + lid;
    const bool valid = (wcol < W_OUT);
    const size_t nh_base = ((size_t)n * F_OUT) * (size_t)(H_OUT * W_OUT)
                         + (size_t)h * W_OUT + wcol;

    v8f acc[4];
#pragma unroll
    for (int t = 0; t < 4; ++t) {
        const float* wb = wgt + (size_t)(t * 16 + lid) * 27 + 18; // channel C-1

        v2f A0, A1, A2;
        A0.x = wb[ka];       A0.y = wb[ka + 1];
        A1.x = wb[kb2];      A1.y = wb[kb2 + 1];
        const float w8 = wb[8];
        A2.x = half ? 0.0f : w8;
        A2.y = 0.0f;

        // Bias folded into the accumulator: C[r] = bias[t*16 + half*8 + r]
        const float* bp = bias + t * 16 + half * 8;
        const v4f blo = *(const v4f*)(bp);
        const v4f bhi = *(const v4f*)(bp + 4);
        v8f c_acc;
        c_acc[0] = blo[0]; c_acc[1] = blo[1]; c_acc[2] = blo[2]; c_acc[3] = blo[3];
        c_acc[4] = bhi[0]; c_acc[5] = bhi[1]; c_acc[6] = bhi[2]; c_acc[7] = bhi[3];

        // D = A(16x4) * B(4x16) + C   (full fp32, RNE)
        c_acc = __builtin_amdgcn_wmma_f32_16x16x4_f32(
            false, A0, false, B0, (short)0, c_acc, false, false);
        c_acc = __builtin_amdgcn_wmma_f32_16x16x4_f32(
            false, A1, false, B1, (short)0, c_acc, false, false);
        c_acc = __builtin_amdgcn_wmma_f32_16x16x4_f32(
            false, A2, false, B2, (short)0, c_acc, false, false);
        acc[t] = c_acc;
    }

    // ---- masked nontemporal stores (single exec toggle) -------------------
    // D layout: VGPR r, lane -> filter f = t*16 + half*8 + r, w col = w0+lid;
    // lanes 0..15 of each half write 16 consecutive floats (64B segments).
    if (valid) {
#pragma unroll
        for (int t = 0; t < 4; ++t) {
#pragma unroll
            for (int r = 0; r < 8; ++r) {
                const int f = t * 16 + half * 8 + r;
                __builtin_nontemporal_store(
                    acc[t][r], out + nh_base + (size_t)f * (H_OUT * W_OUT));
            }
        }
    }
}

extern "C" void kernel_launch(void* const* d_in, const int* in_sizes, int n_in,
                              void* d_out, int out_size, void* d_ws, size_t ws_size,
                              hipStream_t stream) {
    const float* x    = (const float*)d_in[0];
    const float* wgt  = (const float*)d_in[1];
    const float* bias = (const float*)d_in[2];
    float* out = (float*)d_out;

    // 16 imgs * 254 rows * 16 w-tiles = 65024 waves; 8 waves per 256-thr block.
    const int total_waves = N_IMG * H_OUT * WTILES;
    const int blocks = total_waves / WAVES_PER_BLOCK;   // 8128, exact
    conv2dcq_wmma_kernel<<<blocks, 256, 0, stream>>>(x, wgt, bias, out);
}